// FeatureTemporalAttention_33698313405044
// MI455X (gfx1250) — compile-verified
//
#include <hip/hip_runtime.h>
#include <hip/hip_bf16.h>

// ---------------------------------------------------------------------------
// FeatureTemporalAttention for MI455X (gfx1250)
//   B=8, T=128, D=512  ->  M = B*T = 1024 tokens
//   Projections: fp32 WMMA (V_WMMA_F32_16X16X4_F32), LDS-staged tiles,
//                16x32 register-blocked wave tiles, fragment-contiguous LDS
//                layouts so A/B frags are single ds_load_b64 (no v_mov fixup).
//   Temporal branch: per-token streaming softmax (never materializes [D,D]),
//                exp2-folded v_exp_f32, float4 LDS reads, dual-row j-sweep.
// ---------------------------------------------------------------------------

typedef float v2f __attribute__((ext_vector_type(2)));
typedef float v8f __attribute__((ext_vector_type(8)));

constexpr int Bsz = 8;
constexpr int Tsz = 128;
constexpr int D   = 512;
constexpr int M   = Bsz * Tsz;   // 1024 rows

// GEMM tiling: block tile 64x64, 8 waves as 4(M) x 2(N), wave tile 16x32
constexpr int TM = 64;
constexpr int TN = 64;
constexpr int TK = 32;
constexpr int KP = TK + 2;   // padded row stride (8B-aligned rows, no bank conflicts)

struct GemmParams {
    const float* W[6];
    const float* bias[6];
};

// ---------------------------------------------------------------------------
// Kernel 1: ws[z] = x @ W[z]^T + b[z]  for z = 0..5
//   x  : [M, D] row-major;  W : [D(out), D(in)] row-major (nn.Linear)
//   Both LDS tiles keep K innermost so every WMMA fragment (K-pair) is a
//   contiguous, 8B-aligned ds_load_b64.
// ---------------------------------------------------------------------------
__global__ __launch_bounds__(256)
void proj_gemm_wmma(const float* __restrict__ x, GemmParams p,
                    float* __restrict__ ws) {
    const int mBase = blockIdx.x * TM;
    const int nBase = blockIdx.y * TN;
    const int z     = blockIdx.z;

    const float* __restrict__ Wz = p.W[z];
    const float* __restrict__ bz = p.bias[z];
    float* __restrict__ outz = ws + (size_t)z * M * D;

    __shared__ __align__(16) float Ash[TM][KP];   // [m][k]
    __shared__ __align__(16) float Bsh[TN][KP];   // [n][k]  (W tile, K innermost)

    const int tid  = threadIdx.x;
    const int lane = tid & 31;
    const int wid  = tid >> 5;
    const int wm   = wid >> 1;      // 0..3  (wave tile row: 16 rows each)
    const int wn   = wid & 1;       // 0..1  (wave tile col: 32 cols each)
    const int h    = lane >> 4;     // lane half: 0/1
    const int l16  = lane & 15;

    v8f c0 = {};    // cols [wn*32 .. +15]
    v8f c1 = {};    // cols [wn*32+16 .. +31]

    for (int k0 = 0; k0 < D; k0 += TK) {
        // ---- stage A: 64x32 = 2048 elems, 2 x float4 per thread ----
        #pragma unroll
        for (int r = 0; r < 2; ++r) {
            const int q  = tid + r * 256;     // 0..511
            const int am = q >> 3;            // 0..63
            const int ak = (q & 7) << 2;      // 0,4,...,28
            const float4 av = *reinterpret_cast<const float4*>(
                x + (size_t)(mBase + am) * D + k0 + ak);
            Ash[am][ak + 0] = av.x;
            Ash[am][ak + 1] = av.y;
            Ash[am][ak + 2] = av.z;
            Ash[am][ak + 3] = av.w;
        }
        // ---- stage B: Bsh[n][k] = W[nBase+n][k0+k]  (K innermost) ----
        #pragma unroll
        for (int r = 0; r < 2; ++r) {
            const int q  = tid + r * 256;     // 0..511
            const int bn = q >> 3;            // 0..63
            const int bk = (q & 7) << 2;      // 0,4,...,28
            const float4 wv = *reinterpret_cast<const float4*>(
                Wz + (size_t)(nBase + bn) * D + k0 + bk);
            Bsh[bn][bk + 0] = wv.x;
            Bsh[bn][bk + 1] = wv.y;
            Bsh[bn][bk + 2] = wv.z;
            Bsh[bn][bk + 3] = wv.w;
        }
        __syncthreads();

        // ---- 8 k-steps x 2 WMMAs, A fragment reused across both N tiles ----
        const int mf  = wm * 16 + l16;
        const int nf0 = wn * 32 + l16;
        const int nf1 = nf0 + 16;
        #pragma unroll
        for (int kk = 0; kk < TK; kk += 4) {
            // A 16x4: lane L -> M=L%16, VGPR v -> K = 2*(L/16)+v
            // B 4x16: lane L -> N=L%16, VGPR v -> K = 2*(L/16)+v
            // each fragment = contiguous K-pair -> one ds_load_b64
            const v2f a  = *reinterpret_cast<const v2f*>(&Ash[mf][kk + 2 * h]);
            const v2f b0 = *reinterpret_cast<const v2f*>(&Bsh[nf0][kk + 2 * h]);
            const v2f b1 = *reinterpret_cast<const v2f*>(&Bsh[nf1][kk + 2 * h]);
            c0 = __builtin_amdgcn_wmma_f32_16x16x4_f32(
                     false, a, false, b0, (short)0, c0, false, false);
            c1 = __builtin_amdgcn_wmma_f32_16x16x4_f32(
                     false, a, false, b1, (short)0, c1, false, false);
        }
        __syncthreads();
    }

    // ---- writeback + bias.  C/D: lane L, VGPR r -> M=r+8*(L/16), N=L%16 ----
    const int col0 = nBase + wn * 32 + l16;
    const int col1 = col0 + 16;
    const float bv0 = bz[col0];
    const float bv1 = bz[col1];
    #pragma unroll
    for (int r = 0; r < 8; ++r) {
        const int row = mBase + wm * 16 + 8 * h + r;
        outz[(size_t)row * D + col0] = c0[r] + bv0;
        outz[(size_t)row * D + col1] = c1[r] + bv1;
    }
}

// ---------------------------------------------------------------------------
// Block-wide reduction over 256 threads
// ---------------------------------------------------------------------------
struct MaxOp { __device__ float operator()(float a, float b) const { return fmaxf(a, b); } };
struct MinOp { __device__ float operator()(float a, float b) const { return fminf(a, b); } };
struct AddOp { __device__ float operator()(float a, float b) const { return a + b; } };

template <typename Op>
__device__ inline float block_reduce256(float v, float* red, Op op) {
    const int tid = threadIdx.x;
    red[tid] = v;
    __syncthreads();
    #pragma unroll
    for (int s = 128; s > 0; s >>= 1) {
        if (tid < s) red[tid] = op(red[tid], red[tid + s]);
        __syncthreads();
    }
    const float r = red[0];
    __syncthreads();
    return r;
}

// ---------------------------------------------------------------------------
// Kernel 2: per-token fused feature softmax + streaming temporal attention.
//   ws layout: [Qf, Kf, Vf, Qt, Kt, Vt], each [M, D]
//   ctx_f[i] = softmax_i(Qf*Kf) * Vf[i]
//   ctx_t[i] = (sum_j 2^{s2*K_j - m2} V_j) / (sum_j 2^{s2*K_j - m2})
//     with s2 = Qt[i]*log2e, m2 = max(Qt[i]*maxK, Qt[i]*minK)*log2e
//     (== exp(s*K_j - m) up to rounding; exact row max used)
// ---------------------------------------------------------------------------
__global__ __launch_bounds__(256)
void attn_combine(const float* __restrict__ ws, float* __restrict__ out) {
    const int t   = blockIdx.x;     // token 0..1023
    const int tid = threadIdx.x;

    const float* __restrict__ Qf = ws + (size_t)0 * M * D + (size_t)t * D;
    const float* __restrict__ Kf = ws + (size_t)1 * M * D + (size_t)t * D;
    const float* __restrict__ Vf = ws + (size_t)2 * M * D + (size_t)t * D;
    const float* __restrict__ Qt = ws + (size_t)3 * M * D + (size_t)t * D;
    const float* __restrict__ Kt = ws + (size_t)4 * M * D + (size_t)t * D;
    const float* __restrict__ Vt = ws + (size_t)5 * M * D + (size_t)t * D;

    __shared__ __align__(16) float sK[D];
    __shared__ __align__(16) float sV[D];
    __shared__ float red[256];

    const int i0 = tid;
    const int i1 = tid + 256;

    // ---- feature branch: softmax over D of Qf*Kf, scaled by Vf ----
    const float f0 = Qf[i0] * Kf[i0];
    const float f1 = Qf[i1] * Kf[i1];
    const float fmx = block_reduce256(fmaxf(f0, f1), red, MaxOp());
    const float e0 = __expf(f0 - fmx);
    const float e1 = __expf(f1 - fmx);
    const float fsum = block_reduce256(e0 + e1, red, AddOp());
    const float finv = 1.0f / fsum;
    const float cf0 = e0 * finv * Vf[i0];
    const float cf1 = e1 * finv * Vf[i1];

    // ---- temporal branch: stage K,V rows in LDS ----
    const float k0v = Kt[i0];
    const float k1v = Kt[i1];
    sK[i0] = k0v;  sK[i1] = k1v;
    sV[i0] = Vt[i0];  sV[i1] = Vt[i1];
    __syncthreads();

    const float kmax = block_reduce256(fmaxf(k0v, k1v), red, MaxOp());
    const float kmin = block_reduce256(fminf(k0v, k1v), red, MinOp());

    // fold log2(e) so the inner loop is fma -> v_exp_f32 -> fma/add only
    constexpr float L2E = 1.44269504088896340736f;
    const float s0 = Qt[i0];
    const float s1 = Qt[i1];
    const float s2_0 = s0 * L2E;
    const float s2_1 = s1 * L2E;
    const float m2_0 = fmaxf(s0 * kmax, s0 * kmin) * L2E;  // exact row max (log2)
    const float m2_1 = fmaxf(s1 * kmax, s1 * kmin) * L2E;

    float num0 = 0.0f, den0 = 0.0f;
    float num1 = 0.0f, den1 = 0.0f;

    const float4* __restrict__ K4 = reinterpret_cast<const float4*>(sK);
    const float4* __restrict__ V4 = reinterpret_cast<const float4*>(sV);

    #pragma unroll 2
    for (int j4 = 0; j4 < D / 4; ++j4) {
        const float4 k = K4[j4];        // ds_load_b128
        const float4 v = V4[j4];        // ds_load_b128
        float e;
        // row 0 (4 independent exp chains)
        e = __builtin_amdgcn_exp2f(__fmaf_rn(s2_0, k.x, -m2_0));
        num0 = __fmaf_rn(e, v.x, num0); den0 += e;
        e = __builtin_amdgcn_exp2f(__fmaf_rn(s2_0, k.y, -m2_0));
        num0 = __fmaf_rn(e, v.y, num0); den0 += e;
        e = __builtin_amdgcn_exp2f(__fmaf_rn(s2_0, k.z, -m2_0));
        num0 = __fmaf_rn(e, v.z, num0); den0 += e;
        e = __builtin_amdgcn_exp2f(__fmaf_rn(s2_0, k.w, -m2_0));
        num0 = __fmaf_rn(e, v.w, num0); den0 += e;
        // row 1
        e = __builtin_amdgcn_exp2f(__fmaf_rn(s2_1, k.x, -m2_1));
        num1 = __fmaf_rn(e, v.x, num1); den1 += e;
        e = __builtin_amdgcn_exp2f(__fmaf_rn(s2_1, k.y, -m2_1));
        num1 = __fmaf_rn(e, v.y, num1); den1 += e;
        e = __builtin_amdgcn_exp2f(__fmaf_rn(s2_1, k.z, -m2_1));
        num1 = __fmaf_rn(e, v.z, num1); den1 += e;
        e = __builtin_amdgcn_exp2f(__fmaf_rn(s2_1, k.w, -m2_1));
        num1 = __fmaf_rn(e, v.w, num1); den1 += e;
    }

    out[(size_t)t * D + i0] = cf0 + num0 / den0;
    out[(size_t)t * D + i1] = cf1 + num1 / den1;
}

// ---------------------------------------------------------------------------
// Host-side launcher (graph-capture safe: launches only, all on `stream`).
// Inputs: x, then (W_qf,b_qf),(W_kf,b_kf),(W_vf,b_vf),(W_qt,b_qt),(W_kt,b_kt),(W_vt,b_vt)
// Workspace: 6 * M * D floats = 12 MB of d_ws.
// ---------------------------------------------------------------------------
extern "C" void kernel_launch(void* const* d_in, const int* in_sizes, int n_in,
                              void* d_out, int out_size, void* d_ws, size_t ws_size,
                              hipStream_t stream) {
    (void)in_sizes; (void)n_in; (void)out_size; (void)ws_size;

    const float* x = (const float*)d_in[0];
    GemmParams p;
    for (int i = 0; i < 6; ++i) {
        p.W[i]    = (const float*)d_in[1 + 2 * i];
        p.bias[i] = (const float*)d_in[2 + 2 * i];
    }
    float* wsf = (float*)d_ws;
    float* out = (float*)d_out;

    // 6 fused projections: grid = (M/TM, D/TN, 6) = (16, 8, 6)
    proj_gemm_wmma<<<dim3(M / TM, D / TN, 6), 256, 0, stream>>>(x, p, wsf);
    // fused feature+temporal attention, one block per token
    attn_combine<<<M, 256, 0, stream>>>(wsf, out);
}